// MambaMIL_2D_37340445671872
// MI455X (gfx1250) — compile-verified
//
#include <hip/hip_runtime.h>
#include <hip/hip_bf16.h>
#include <math.h>

// ---------------------------------------------------------------------------
// MambaMIL_2D forward for MI455X (gfx1250, wave32).
// GEMMs: V_WMMA_F32_16X16X4_F32, 32x32 register-blocked wave tile
// (2 A-frags x 2 B-frags -> 4 independent WMMA accumulator chains,
//  8 FLOP per byte of L2 traffic, model is fully L2-resident).
// ---------------------------------------------------------------------------

typedef __attribute__((ext_vector_type(2))) float v2f;
typedef __attribute__((ext_vector_type(8))) float v8f;

#define LQ 2048      // sequence length
#define DM 512       // d_model
#define DI 1024      // d_inner
#define DS 16        // d_state
#define DTR 32       // dt_rank

// ---------------------------------------------------------------------------
// WMMA fp32 GEMM:  C[M,N] = act(A[M,K] @ B[K,N] + bias) + res
// One 32x32 C block per wave (four 16x16 WMMA tiles). M,N mult of 32, K of 4.
// act: 0=none 1=gelu(exact erf) 2=silu 3=tanh 4=softplus
// ---------------------------------------------------------------------------
__launch_bounds__(128)
__global__ void gemm_wmma_f32(const float* __restrict__ A, const float* __restrict__ B,
                              const float* __restrict__ bias, const float* res,
                              float* C, int M, int N, int K,
                              int lda, int ldb, int ldc, int act)
{
    const int lane = threadIdx.x & 31;
    const int wave = threadIdx.x >> 5;
    const int ntn  = N >> 5;                 // 32-wide column blocks
    const int ntm  = M >> 5;
    int tile = blockIdx.x * (blockDim.x >> 5) + wave;
    if (tile >= ntm * ntn) return;           // wave-uniform exit (EXEC stays full)
    const int tm = tile / ntn;
    const int tn = tile - tm * ntn;

    const int row  = lane & 15;
    const int half = lane >> 4;              // 0: K base+0/1, 1: K base+2/3
    const float* Arow0 = A + (size_t)(tm * 32 + row)      * lda;
    const float* Arow1 = A + (size_t)(tm * 32 + 16 + row) * lda;
    const float* Bcol0 = B + (tn * 32 + row);        // 'row' = N-column in B frag
    const float* Bcol1 = B + (tn * 32 + 16 + row);

    v8f acc00 = {}, acc01 = {}, acc10 = {}, acc11 = {};
    for (int k = 0; k < K; k += 4) {
        if ((k & 31) == 0) {
            __builtin_prefetch(Arow0 + k + 64, 0, 1);            // global_prefetch_b8
            __builtin_prefetch(Bcol0 + (size_t)(k + 16) * ldb, 0, 1);
        }
        const int kk = k + half * 2;
        v2f a0, a1, b0, b1;
        a0.x = Arow0[kk];  a0.y = Arow0[kk + 1];
        a1.x = Arow1[kk];  a1.y = Arow1[kk + 1];
        b0.x = Bcol0[(size_t)kk * ldb];  b0.y = Bcol0[(size_t)(kk + 1) * ldb];
        b1.x = Bcol1[(size_t)kk * ldb];  b1.y = Bcol1[(size_t)(kk + 1) * ldb];
        // four independent accumulator chains -> matrix-pipe ILP
        acc00 = __builtin_amdgcn_wmma_f32_16x16x4_f32(false, a0, false, b0, (short)0, acc00, false, false);
        acc01 = __builtin_amdgcn_wmma_f32_16x16x4_f32(false, a0, false, b1, (short)0, acc01, false, false);
        acc10 = __builtin_amdgcn_wmma_f32_16x16x4_f32(false, a1, false, b0, (short)0, acc10, false, false);
        acc11 = __builtin_amdgcn_wmma_f32_16x16x4_f32(false, a1, false, b1, (short)0, acc11, false, false);
    }

    const int col0 = tn * 32 + (lane & 15);
    const int col1 = col0 + 16;
    const float bv0 = bias ? bias[col0] : 0.0f;
    const float bv1 = bias ? bias[col1] : 0.0f;

#pragma unroll
    for (int q = 0; q < 4; ++q) {            // q = {i,j} tile within 32x32 block
        const int i = q >> 1, j = q & 1;
        const v8f accq = (q == 0) ? acc00 : (q == 1) ? acc01 : (q == 2) ? acc10 : acc11;
        const int col  = j ? col1 : col0;
        const float bv = j ? bv1 : bv0;
#pragma unroll
        for (int r = 0; r < 8; ++r) {
            int rr = tm * 32 + i * 16 + r + 8 * half;  // documented C layout
            float v = accq[r] + bv;
            if (act == 1)      v = 0.5f * v * (1.0f + erff(v * 0.70710678118654752f));
            else if (act == 2) v = v / (1.0f + expf(-v));
            else if (act == 3) v = tanhf(v);
            else if (act == 4) v = (v > 20.0f) ? v : log1pf(expf(v));
            size_t off = (size_t)rr * ldc + col;
            if (res) v += res[off];
            C[off] = v;
        }
    }
}

// --- pos-emb: h += coords @ W_pos + b_pos  (K=2, elementwise) ---------------
__global__ void posemb_add(float* h, const float* __restrict__ coords,
                           const float* __restrict__ Wp, const float* __restrict__ bp)
{
    int idx = blockIdx.x * blockDim.x + threadIdx.x;
    if (idx >= LQ * DM) return;
    int t = idx >> 9, d = idx & (DM - 1);
    h[idx] += coords[t * 2] * Wp[d] + coords[t * 2 + 1] * Wp[DM + d] + bp[d];
}

// --- RMSNorm over d_model, one block per row --------------------------------
__global__ void rmsnorm_k(const float* __restrict__ h, const float* __restrict__ w,
                          float* __restrict__ out)
{
    __shared__ float red[8];
    __shared__ float sinv;
    int t = blockIdx.x, tid = threadIdx.x;
    const float* row = h + (size_t)t * DM;
    float s = 0.f;
    for (int i = tid; i < DM; i += 256) { float v = row[i]; s += v * v; }
    for (int off = 16; off; off >>= 1) s += __shfl_xor(s, off, 32);
    if ((tid & 31) == 0) red[tid >> 5] = s;
    __syncthreads();
    if (tid == 0) {
        float tot = 0.f;
        for (int i = 0; i < 8; ++i) tot += red[i];
        sinv = rsqrtf(tot / (float)DM + 1e-5f);
    }
    __syncthreads();
    for (int i = tid; i < DM; i += 256) out[(size_t)t * DM + i] = row[i] * sinv * w[i];
}

// --- LayerNorm over d_model, one block per row ------------------------------
__global__ void layernorm_k(const float* __restrict__ h, const float* __restrict__ w,
                            const float* __restrict__ b, float* __restrict__ out)
{
    __shared__ float red1[8], red2[8];
    __shared__ float smean, sinv;
    int t = blockIdx.x, tid = threadIdx.x;
    const float* row = h + (size_t)t * DM;
    float s = 0.f, s2 = 0.f;
    for (int i = tid; i < DM; i += 256) { float v = row[i]; s += v; s2 += v * v; }
    for (int off = 16; off; off >>= 1) {
        s  += __shfl_xor(s,  off, 32);
        s2 += __shfl_xor(s2, off, 32);
    }
    if ((tid & 31) == 0) { red1[tid >> 5] = s; red2[tid >> 5] = s2; }
    __syncthreads();
    if (tid == 0) {
        float ts = 0.f, ts2 = 0.f;
        for (int i = 0; i < 8; ++i) { ts += red1[i]; ts2 += red2[i]; }
        float m = ts / (float)DM;
        smean = m;
        sinv  = rsqrtf(ts2 / (float)DM - m * m + 1e-5f);
    }
    __syncthreads();
    for (int i = tid; i < DM; i += 256)
        out[(size_t)t * DM + i] = (row[i] - smean) * sinv * w[i] + b[i];
}

// --- causal depthwise conv1d (k=4, left pad 3) + SiLU -----------------------
__global__ void dwconv_silu(const float* __restrict__ xz, const float* __restrict__ cw,
                            const float* __restrict__ cb, float* __restrict__ xc)
{
    int idx = blockIdx.x * blockDim.x + threadIdx.x;
    if (idx >= LQ * DI) return;
    int t = idx >> 10, d = idx & (DI - 1);
    float s = cb[d];
#pragma unroll
    for (int k = 0; k < 4; ++k) {
        int tt = t - 3 + k;
        if (tt >= 0) s += xz[(size_t)tt * (2 * DI) + d] * cw[d * 4 + k];
    }
    xc[idx] = s / (1.0f + expf(-s));
}

// --- selective scan: 2 channels per wave, 16 states per half-wave -----------
__global__ void mamba_scan(const float* __restrict__ delta, const float* __restrict__ xc,
                           const float* __restrict__ dbc, const float* __restrict__ xz,
                           const float* __restrict__ alog, const float* __restrict__ Dp,
                           float* __restrict__ ys)
{
    int wave = (blockIdx.x * blockDim.x + threadIdx.x) >> 5;
    int lane = threadIdx.x & 31;
    int half = lane >> 4, n = lane & 15;
    int d = wave * 2 + half;
    if (d >= DI) return;

    const float Adn = -expf(alog[d * DS + n]);   // A = -exp(A_log), constant over t
    const float Dd  = Dp[d];
    float hstate = 0.0f;
    for (int t = 0; t < LQ; ++t) {
        float dl = delta[(size_t)t * DI + d];
        float xv = xc[(size_t)t * DI + d];
        float Bn = dbc[(size_t)t * 64 + DTR + n];
        float Cn = dbc[(size_t)t * 64 + DTR + DS + n];
        float dA = expf(dl * Adn);
        hstate = dA * hstate + dl * xv * Bn;
        float contrib = hstate * Cn;
        contrib += __shfl_xor(contrib, 1, 32);   // reduce 16 states in half-wave
        contrib += __shfl_xor(contrib, 2, 32);
        contrib += __shfl_xor(contrib, 4, 32);
        contrib += __shfl_xor(contrib, 8, 32);
        if (n == 0) {
            float zv = xz[(size_t)t * (2 * DI) + DI + d];
            ys[(size_t)t * DI + d] = (contrib + xv * Dd) * (zv / (1.0f + expf(-zv)));
        }
    }
}

// --- attention score: s[t] = a1[t,:] . Wa2 + ba2 ----------------------------
__global__ void attn_score(const float* __restrict__ a1, const float* __restrict__ Wa2,
                           const float* __restrict__ ba2, float* __restrict__ s)
{
    int t = blockIdx.x * blockDim.x + threadIdx.x;
    if (t >= LQ) return;
    float acc = ba2[0];
    for (int i = 0; i < 128; ++i) acc += a1[(size_t)t * 128 + i] * Wa2[i];
    s[t] = acc;
}

// --- softmax over L=2048, single block --------------------------------------
__global__ void softmax2048(const float* __restrict__ s, float* __restrict__ w)
{
    __shared__ float red[8];
    __shared__ float bmax, bsum;
    int tid = threadIdx.x;
    float m = -1e30f;
    for (int i = tid; i < LQ; i += 256) m = fmaxf(m, s[i]);
    for (int off = 16; off; off >>= 1) m = fmaxf(m, __shfl_xor(m, off, 32));
    if ((tid & 31) == 0) red[tid >> 5] = m;
    __syncthreads();
    if (tid == 0) {
        float mm = red[0];
        for (int i = 1; i < 8; ++i) mm = fmaxf(mm, red[i]);
        bmax = mm;
    }
    __syncthreads();
    float sum = 0.f;
    for (int i = tid; i < LQ; i += 256) sum += expf(s[i] - bmax);
    for (int off = 16; off; off >>= 1) sum += __shfl_xor(sum, off, 32);
    if ((tid & 31) == 0) red[tid >> 5] = sum;
    __syncthreads();
    if (tid == 0) {
        float ss = 0.f;
        for (int i = 0; i < 8; ++i) ss += red[i];
        bsum = ss;
    }
    __syncthreads();
    for (int i = tid; i < LQ; i += 256) w[i] = expf(s[i] - bmax) / bsum;
}

// --- pooled[d] = sum_t w[t]*h[t,d] ------------------------------------------
__global__ void pool_k(const float* __restrict__ w, const float* __restrict__ hn,
                       float* __restrict__ pooled)
{
    int d = blockIdx.x * blockDim.x + threadIdx.x;
    if (d >= DM) return;
    float acc = 0.f;
    for (int t = 0; t < LQ; ++t) acc += w[t] * hn[(size_t)t * DM + d];
    pooled[d] = acc;
}

// --- ConvTranspose2d (k=4), PyTorch semantics, weight [Cin,Cout,4,4] --------
__global__ void convt_k(const float* __restrict__ in, const float* __restrict__ wgt,
                        const float* __restrict__ bias, float* __restrict__ out,
                        int Cin, int Cout, int Hin, int Hout, int s, int p, int relu)
{
    int idx = blockIdx.x * blockDim.x + threadIdx.x;
    if (idx >= Cout * Hout * Hout) return;
    int ox = idx % Hout;
    int oy = (idx / Hout) % Hout;
    int co = idx / (Hout * Hout);
    float acc = bias[co];
    for (int ky = 0; ky < 4; ++ky) {
        int iyn = oy + p - ky;
        if (iyn < 0 || (iyn % s)) continue;
        int iy = iyn / s;
        if (iy >= Hin) continue;
        for (int kx = 0; kx < 4; ++kx) {
            int ixn = ox + p - kx;
            if (ixn < 0 || (ixn % s)) continue;
            int ix = ixn / s;
            if (ix >= Hin) continue;
            for (int ci = 0; ci < Cin; ++ci)
                acc += in[((size_t)ci * Hin + iy) * Hin + ix] *
                       wgt[(((size_t)ci * Cout + co) * 4 + ky) * 4 + kx];
        }
    }
    out[idx] = (relu && acc < 0.f) ? 0.f : acc;
}

// ---------------------------------------------------------------------------
static inline void launch_gemm(const float* A, const float* B, const float* bias,
                               const float* res, float* C, int M, int N, int K,
                               int lda, int ldb, int ldc, int act, hipStream_t st)
{
    int tiles = (M / 32) * (N / 32);
    int blocks = (tiles + 3) / 4;          // 4 waves (32x32 blocks) per 128-thread block
    gemm_wmma_f32<<<blocks, 128, 0, st>>>(A, B, bias, res, C, M, N, K, lda, ldb, ldc, act);
}

extern "C" void kernel_launch(void* const* d_in, const int* in_sizes, int n_in,
                              void* d_out, int out_size, void* d_ws, size_t ws_size,
                              hipStream_t stream)
{
    const float* x       = (const float*)d_in[0];
    const float* coords  = (const float*)d_in[1];
    const float* W_fc1   = (const float*)d_in[2];
    const float* b_fc1   = (const float*)d_in[3];
    const float* W_pos   = (const float*)d_in[4];
    const float* b_pos   = (const float*)d_in[5];
    const float* in_w    = (const float*)d_in[6];   // [2,512,2048]
    const float* conv_w  = (const float*)d_in[7];   // [2,1024,1,4]
    const float* conv_b  = (const float*)d_in[8];   // [2,1024]
    const float* x_w     = (const float*)d_in[9];   // [2,1024,64]
    const float* dt_w    = (const float*)d_in[10];  // [2,32,1024]
    const float* dt_b    = (const float*)d_in[11];  // [2,1024]
    const float* alog    = (const float*)d_in[12];  // [2,1024,16]
    const float* Dp      = (const float*)d_in[13];  // [2,1024]
    const float* out_w   = (const float*)d_in[14];  // [2,1024,512]
    const float* rms_w   = (const float*)d_in[15];  // [2,512]
    const float* ln_w    = (const float*)d_in[16];
    const float* ln_b    = (const float*)d_in[17];
    const float* Wa1     = (const float*)d_in[18];  // [512,128]
    const float* ba1     = (const float*)d_in[19];
    const float* Wa2     = (const float*)d_in[20];  // [128,1]
    const float* ba2     = (const float*)d_in[21];
    const float* ct_w[5] = {(const float*)d_in[22], (const float*)d_in[24],
                            (const float*)d_in[26], (const float*)d_in[28],
                            (const float*)d_in[30]};
    const float* ct_b[5] = {(const float*)d_in[23], (const float*)d_in[25],
                            (const float*)d_in[27], (const float*)d_in[29],
                            (const float*)d_in[31]};

    // workspace layout (floats)
    float* ws = (float*)d_ws;
    float* h      = ws;                       // 2048*512
    float* hn     = h      + (size_t)LQ * DM; // 2048*512
    float* xz     = hn     + (size_t)LQ * DM; // 2048*2048
    float* xc     = xz     + (size_t)LQ * 2 * DI; // 2048*1024
    float* dbc    = xc     + (size_t)LQ * DI; // 2048*64
    float* delta  = dbc    + (size_t)LQ * 64; // 2048*1024
    float* ys     = delta  + (size_t)LQ * DI; // 2048*1024
    float* a1     = ys     + (size_t)LQ * DI; // 2048*128
    float* scores = a1     + (size_t)LQ * 128; // 2048
    float* attw   = scores + LQ;               // 2048
    float* pooled = attw   + LQ;               // 512
    float* cbuf0  = pooled + DM;               // 131072
    float* cbuf1  = cbuf0  + 131072;           // 131072

    // fc1: h = gelu(x @ W_fc1 + b_fc1), then h += coords @ W_pos + b_pos
    launch_gemm(x, W_fc1, b_fc1, nullptr, h, LQ, DM, 1024, 1024, DM, DM, 1, stream);
    posemb_add<<<(LQ * DM) / 256, 256, 0, stream>>>(h, coords, W_pos, b_pos);

    for (int l = 0; l < 2; ++l) {
        // pre-norm
        rmsnorm_k<<<LQ, 256, 0, stream>>>(h, rms_w + (size_t)l * DM, hn);
        // in_proj -> xz [L, 2*DI]
        launch_gemm(hn, in_w + (size_t)l * DM * 2 * DI, nullptr, nullptr,
                    xz, LQ, 2 * DI, DM, DM, 2 * DI, 2 * DI, 0, stream);
        // depthwise causal conv + silu -> xc
        dwconv_silu<<<(LQ * DI) / 256, 256, 0, stream>>>(
            xz, conv_w + (size_t)l * DI * 4, conv_b + (size_t)l * DI, xc);
        // x_proj -> dbc [L, 64] = [dt(32) | B(16) | C(16)]
        launch_gemm(xc, x_w + (size_t)l * DI * 64, nullptr, nullptr,
                    dbc, LQ, 64, DI, DI, 64, 64, 0, stream);
        // dt_proj + softplus -> delta [L, DI]   (A = dbc[:, :32], lda=64)
        launch_gemm(dbc, dt_w + (size_t)l * DTR * DI, dt_b + (size_t)l * DI, nullptr,
                    delta, LQ, DI, DTR, 64, DI, DI, 4, stream);
        // selective scan + D-skip + silu(z) gate -> ys
        mamba_scan<<<64, 256, 0, stream>>>(delta, xc, dbc, xz,
                                           alog + (size_t)l * DI * DS,
                                           Dp + (size_t)l * DI, ys);
        // out_proj + residual: h = h + ys @ out_w
        launch_gemm(ys, out_w + (size_t)l * DI * DM, nullptr, h,
                    h, LQ, DM, DI, DI, DM, DM, 0, stream);
    }

    // final LayerNorm -> hn
    layernorm_k<<<LQ, 256, 0, stream>>>(h, ln_w, ln_b, hn);

    // attention pooling
    launch_gemm(hn, Wa1, ba1, nullptr, a1, LQ, 128, DM, DM, 128, 128, 3, stream);
    attn_score<<<LQ / 256, 256, 0, stream>>>(a1, Wa2, ba2, scores);
    softmax2048<<<1, 256, 0, stream>>>(scores, attw);
    pool_k<<<2, 256, 0, stream>>>(attw, hn, pooled);

    // ConvTranspose2d decoder: 1 -> 4 -> 16 -> 32 -> 64 -> 128
    convt_k<<<(256 * 4 * 4 + 255) / 256, 256, 0, stream>>>(
        pooled, ct_w[0], ct_b[0], cbuf0, 512, 256, 1, 4, 1, 0, 1);
    convt_k<<<(128 * 16 * 16 + 255) / 256, 256, 0, stream>>>(
        cbuf0, ct_w[1], ct_b[1], cbuf1, 256, 128, 4, 16, 4, 0, 1);
    convt_k<<<(64 * 32 * 32 + 255) / 256, 256, 0, stream>>>(
        cbuf1, ct_w[2], ct_b[2], cbuf0, 128, 64, 16, 32, 2, 1, 1);
    convt_k<<<(32 * 64 * 64 + 255) / 256, 256, 0, stream>>>(
        cbuf0, ct_w[3], ct_b[3], cbuf1, 64, 32, 32, 64, 2, 1, 1);
    convt_k<<<(11 * 128 * 128 + 255) / 256, 256, 0, stream>>>(
        cbuf1, ct_w[4], ct_b[4], (float*)d_out, 32, 11, 64, 128, 2, 1, 0);
}